// GraphPropagator_85624468013618
// MI455X (gfx1250) — compile-verified
//
#include <hip/hip_runtime.h>
#include <hip/hip_bf16.h>

// Problem constants (match reference)
#define Bc 4
#define Nc 50000
#define Gc 2
#define Ec 500000
#define Hc 64
#define Rc 16

typedef __attribute__((ext_vector_type(2))) float v2f;
typedef __attribute__((ext_vector_type(8))) float v8f;

// ---- workspace layout (floats) ----
// s   : [G*B*N]            scatter accumulators  (400000)
// t   : [G*N]              weight-sum per dst    (100000)
// acc : [G*B*H]            relu-sum accumulators (512)
// pq  : [2*H]              p = lin_w@post_w, q = lin_b@post_w (128)
#define WS_S   0
#define WS_T   (WS_S + Gc*Bc*Nc)
#define WS_ACC (WS_T + Gc*Nc)
#define WS_PQ  (WS_ACC + Gc*Bc*Hc)
#define WS_TOT (WS_PQ + 2*Hc)

__device__ __forceinline__ float relu1(float x) {
    // single v_med3_f32: median{x, 0, +inf} == relu(x) for finite x
    return __builtin_amdgcn_fmed3f(x, 0.0f, __builtin_inff());
}

// ------------------------------------------------------------------
// 0) zero the accumulated regions (must re-zero every call)
// ------------------------------------------------------------------
__global__ void gp_zero_kernel(float* ws, int count) {
    int i = blockIdx.x * blockDim.x + threadIdx.x;
    if (i < count) ws[i] = 0.0f;
}

// ------------------------------------------------------------------
// 1) p[h] = sum_k lin_w[k]*post_w[k,h];  q[h] = sum_k lin_b[k]*post_w[k,h]
// ------------------------------------------------------------------
__global__ void gp_pq_kernel(const float* __restrict__ lin_w,
                             const float* __restrict__ lin_b,
                             const float* __restrict__ post_w,
                             float* __restrict__ pq) {
    int h = threadIdx.x;
    if (h >= Hc) return;
    float p = 0.f, q = 0.f;
    #pragma unroll 4
    for (int k = 0; k < Hc; ++k) {
        float w = post_w[k * Hc + h];
        p = fmaf(lin_w[k], w, p);
        q = fmaf(lin_b[k], w, q);
    }
    pq[h]      = p;
    pq[Hc + h] = q;
}

// ------------------------------------------------------------------
// 2) Edge kernel: gate = sigmoid(<u[src], v[dst]>), w = ew * gate
//    t[g,dst]   += w
//    s[g,b,dst] += w * mask[b,src]   (b = 0..3)
//    Edge stream (12MB, touched once) loaded non-temporal so the reused
//    gate tables (6.4MB) + s/t atomic targets (2MB) stay L2-resident.
// ------------------------------------------------------------------
__global__ void gp_edge_kernel(const float* __restrict__ pert_mask,
                               const int*   __restrict__ edge_src,
                               const int*   __restrict__ edge_dst,
                               const float* __restrict__ edge_weight,
                               const float* __restrict__ gate_u,
                               const float* __restrict__ gate_v,
                               float* __restrict__ s,
                               float* __restrict__ t) {
    int e = blockIdx.x * blockDim.x + threadIdx.x;
    int g = blockIdx.y;
    if (e >= Ec) return;

    size_t ge  = (size_t)g * Ec + e;
    int src = __builtin_nontemporal_load(&edge_src[ge]);
    int dst = __builtin_nontemporal_load(&edge_dst[ge]);
    float ew = __builtin_nontemporal_load(&edge_weight[ge]);

    const float4* u4 = reinterpret_cast<const float4*>(gate_u + ((size_t)g * Nc + src) * Rc);
    const float4* w4 = reinterpret_cast<const float4*>(gate_v + ((size_t)g * Nc + dst) * Rc);
    float dot = 0.f;
    #pragma unroll
    for (int k = 0; k < Rc / 4; ++k) {
        float4 a = u4[k];
        float4 b = w4[k];
        dot = fmaf(a.x, b.x, dot);
        dot = fmaf(a.y, b.y, dot);
        dot = fmaf(a.z, b.z, dot);
        dot = fmaf(a.w, b.w, dot);
    }
    float gate = 1.0f / (1.0f + __expf(-dot));
    float w = ew * gate;

    atomicAdd(&t[(size_t)g * Nc + dst], w);
    #pragma unroll
    for (int b = 0; b < Bc; ++b) {
        float m = pert_mask[(size_t)b * Nc + src];
        atomicAdd(&s[((size_t)(g * Bc + b)) * Nc + dst], w * m);
    }
}

// ------------------------------------------------------------------
// 3) WMMA relu-mean: for each (g,b), acc[g,b,h] = sum_n relu(s*p + t*q + c)
//    One V_WMMA_F32_16X16X4_F32 handles 16 nodes x 16 channels:
//      A (16x4): lanes 0-15 -> (K0=s[n], K1=t[n]); lanes 16-31 -> (K2=1, K3=0)
//      B (4x16): VGPR0: lanes0-15 K0=p[col], lanes16-31 K2=post_b[col]
//                VGPR1: lanes0-15 K1=q[col], lanes16-31 K3=0
//    ReLU (v_med3) + accumulate on the 8 D VGPRs, summed across chunks.
// ------------------------------------------------------------------
__global__ void gp_relu_mean_wmma(const float* __restrict__ s,
                                  const float* __restrict__ t,
                                  const float* __restrict__ pq,
                                  const float* __restrict__ post_b,
                                  float* __restrict__ acc) {
    const int lane   = threadIdx.x & 31;
    const int wave   = threadIdx.x >> 5;
    const int gb     = blockIdx.y;          // g*B + b
    const int g      = gb / Bc;
    const int waveId = blockIdx.x * (blockDim.x >> 5) + wave;
    const int nWaves = gridDim.x * (blockDim.x >> 5);

    const float* s_gb = s + (size_t)gb * Nc;
    const float* t_g  = t + (size_t)g * Nc;

    const int col = lane & 15;
    const int hi  = lane >> 4;   // 0: lanes 0-15, 1: lanes 16-31

    // Build the 4 column-block B operands (uniform across the loop).
    v2f Bv[Hc / 16];
    #pragma unroll
    for (int cb = 0; cb < Hc / 16; ++cb) {
        float b0 = hi ? post_b[cb * 16 + col] : pq[cb * 16 + col];        // K0=p / K2=bias
        float b1 = hi ? 0.0f                  : pq[Hc + cb * 16 + col];   // K1=q / K3=0
        Bv[cb][0] = b0;
        Bv[cb][1] = b1;
    }

    v8f racc[Hc / 16];
    #pragma unroll
    for (int cb = 0; cb < Hc / 16; ++cb) racc[cb] = (v8f){0.f,0.f,0.f,0.f,0.f,0.f,0.f,0.f};

    const int nChunks = Nc / 16;   // 3125, exact
    for (int chunk = waveId; chunk < nChunks; chunk += nWaves) {
        int n0 = chunk * 16;
        // lane<16: real (s,t) rows; lane>=16: constant (1,0) rows feeding bias.
        float a0 = (hi == 0) ? s_gb[n0 + col] : 1.0f;
        float a1 = (hi == 0) ? t_g[n0 + col]  : 0.0f;
        v2f A; A[0] = a0; A[1] = a1;

        #pragma unroll
        for (int cb = 0; cb < Hc / 16; ++cb) {
            v8f czero = (v8f){0.f,0.f,0.f,0.f,0.f,0.f,0.f,0.f};
            v8f d = __builtin_amdgcn_wmma_f32_16x16x4_f32(
                false, A, false, Bv[cb], (short)0, czero, false, false);
            #pragma unroll
            for (int j = 0; j < 8; ++j)
                racc[cb][j] += relu1(d[j]);
        }
    }

    // Each lane holds partial sums for channel (cb*16 + col) over its row-halves.
    #pragma unroll
    for (int cb = 0; cb < Hc / 16; ++cb) {
        float sum = racc[cb][0] + racc[cb][1] + racc[cb][2] + racc[cb][3]
                  + racc[cb][4] + racc[cb][5] + racc[cb][6] + racc[cb][7];
        atomicAdd(&acc[(size_t)gb * Hc + cb * 16 + col], sum);
    }
}

// ------------------------------------------------------------------
// 4) Softmax mixing: out[b,h] = sum_g softmax_g(ctx@mix_w + mix_b) * acc[g,b,h]/N
// ------------------------------------------------------------------
__global__ void gp_mix_kernel(const float* __restrict__ ctx_emb,
                              const float* __restrict__ mix_w,
                              const float* __restrict__ mix_b,
                              const float* __restrict__ acc,
                              float* __restrict__ out) {
    int tid = blockIdx.x * blockDim.x + threadIdx.x;
    if (tid >= Bc * Hc) return;
    int b = tid / Hc;
    int h = tid % Hc;

    float logit[Gc];
    #pragma unroll
    for (int g = 0; g < Gc; ++g) {
        float acc_l = mix_b[g];
        for (int k = 0; k < Hc; ++k)
            acc_l = fmaf(ctx_emb[b * Hc + k], mix_w[k * Gc + g], acc_l);
        logit[g] = acc_l;
    }
    float m = fmaxf(logit[0], logit[1]);
    float e0 = __expf(logit[0] - m);
    float e1 = __expf(logit[1] - m);
    float inv = 1.0f / (e0 + e1);
    float w0 = e0 * inv, w1 = e1 * inv;

    const float invN = 1.0f / (float)Nc;
    float r = w0 * acc[(size_t)(0 * Bc + b) * Hc + h] * invN
            + w1 * acc[(size_t)(1 * Bc + b) * Hc + h] * invN;
    out[(size_t)b * Hc + h] = r;
}

// ------------------------------------------------------------------
extern "C" void kernel_launch(void* const* d_in, const int* in_sizes, int n_in,
                              void* d_out, int out_size, void* d_ws, size_t ws_size,
                              hipStream_t stream) {
    const float* pert_mask   = (const float*)d_in[0];
    const float* ctx_emb     = (const float*)d_in[1];
    const int*   edge_src    = (const int*)  d_in[2];
    const int*   edge_dst    = (const int*)  d_in[3];
    const float* edge_weight = (const float*)d_in[4];
    const float* gate_u      = (const float*)d_in[5];
    const float* gate_v      = (const float*)d_in[6];
    const float* lin_w       = (const float*)d_in[7];
    const float* lin_b       = (const float*)d_in[8];
    const float* post_w      = (const float*)d_in[9];
    const float* post_b      = (const float*)d_in[10];
    const float* mix_w       = (const float*)d_in[11];
    const float* mix_b       = (const float*)d_in[12];
    float* out = (float*)d_out;

    float* ws  = (float*)d_ws;
    float* s_p   = ws + WS_S;
    float* t_p   = ws + WS_T;
    float* acc_p = ws + WS_ACC;
    float* pq_p  = ws + WS_PQ;

    // 0) zero accumulated regions (s, t, acc)
    {
        int count = WS_PQ;  // s + t + acc are contiguous from 0
        gp_zero_kernel<<<(count + 255) / 256, 256, 0, stream>>>(ws, count);
    }
    // 1) p/q precompute
    gp_pq_kernel<<<1, 64, 0, stream>>>(lin_w, lin_b, post_w, pq_p);
    // 2) edge gather/gate/scatter
    {
        dim3 grid((Ec + 255) / 256, Gc);
        gp_edge_kernel<<<grid, 256, 0, stream>>>(pert_mask, edge_src, edge_dst,
                                                 edge_weight, gate_u, gate_v,
                                                 s_p, t_p);
    }
    // 3) WMMA relu-mean
    {
        dim3 grid(40, Gc * Bc);   // 40 blocks x 4 waves = 160 waves per (g,b)
        gp_relu_mean_wmma<<<grid, 128, 0, stream>>>(s_p, t_p, pq_p, post_b, acc_p);
    }
    // 4) softmax mix -> out [B,H]
    gp_mix_kernel<<<1, Bc * Hc, 0, stream>>>(ctx_emb, mix_w, mix_b, acc_p, out);
}